// AdaptiveComputationTime_85753317032102
// MI455X (gfx1250) — compile-verified
//
#include <hip/hip_runtime.h>
#include <cstdint>
#include <cstddef>

// Problem constants (fixed by the reference).
#define B_  32
#define M_  1024
#define H_  512
#define BM_ (B_ * M_)
#define THRESH 0.99f

typedef __attribute__((ext_vector_type(2))) float v2f;
typedef __attribute__((ext_vector_type(8))) float v8f;

// ---------------------------------------------------------------------------
// Kernel 1: dots[r] = dot(h[r, :], p_w)  for all r in [0, B*M)
// Uses V_WMMA_F32_16X16X4_F32: one wave32 produces 16 row-dots.
//   A (16x4 f32):  lane&15 = M row; lane>=16 half supplies K+2; 2 VGPRs = K,K+1
//   B (4x16 f32):  only output column N==0 is nonzero (= p_w chunk), so the
//                  accumulated C column 0 holds the 16 dot products.
//   C/D: lane 0 VGPR r = (M=r, N=0); lane 16 VGPR r = (M=8+r, N=0)
// ---------------------------------------------------------------------------
__global__ __launch_bounds__(256) void act_gemv_wmma(
    const float* __restrict__ h, const float* __restrict__ p_w,
    float* __restrict__ dots) {
  const int lane  = threadIdx.x & 31;
  const int wid   = (blockIdx.x * 256 + threadIdx.x) >> 5;  // global wave id
  const int row0  = wid * 16;
  const int mrow  = lane & 15;
  const int koff  = (lane >> 4) << 1;                       // 0 or 2
  const bool col0 = (mrow == 0);                            // lanes 0 and 16

  const float* arow = h + (size_t)(row0 + mrow) * H_ + koff;

  v8f c = {0.f, 0.f, 0.f, 0.f, 0.f, 0.f, 0.f, 0.f};
#pragma unroll 8
  for (int k0 = 0; k0 < H_; k0 += 4) {
    const float2 a2 = *(const float2*)(arow + k0);
    const float2 w2 = *(const float2*)(p_w + k0 + koff);
    v2f a; a.x = a2.x;               a.y = a2.y;
    v2f b; b.x = col0 ? w2.x : 0.f;  b.y = col0 ? w2.y : 0.f;
    c = __builtin_amdgcn_wmma_f32_16x16x4_f32(
            /*neg_a=*/false, a, /*neg_b=*/false, b,
            /*c_mod=*/(short)0, c, /*reuse_a=*/false, /*reuse_b=*/false);
  }

  if (lane == 0) {
#pragma unroll
    for (int i = 0; i < 8; ++i) dots[row0 + i] = c[i];
  } else if (lane == 16) {
#pragma unroll
    for (int i = 0; i < 8; ++i) dots[row0 + 8 + i] = c[i];
  }
}

// ---------------------------------------------------------------------------
// Kernel 2: per batch-row scans + halting decision.
// One 1024-thread workgroup per b. LDS Hillis-Steele inclusive scan.
// ---------------------------------------------------------------------------
__device__ __forceinline__ int block_incl_scan(int v, int m, int* sm) {
  sm[m] = v;
  __syncthreads();
#pragma unroll
  for (int off = 1; off < M_; off <<= 1) {
    int t = (m >= off) ? sm[m - off] : 0;
    __syncthreads();
    sm[m] += t;
    __syncthreads();
  }
  int r = sm[m];
  __syncthreads();
  return r;
}

__global__ __launch_bounds__(1024) void act_scan(
    const uint8_t* __restrict__ run, const float* __restrict__ acc_p,
    const float* __restrict__ dots, const float* __restrict__ coeff_p,
    const float* __restrict__ pb_p, int* __restrict__ rank_ws,
    float* __restrict__ upd_ws, int* __restrict__ srcmap,
    int* __restrict__ count_new) {
  __shared__ int sm[M_];
  const int b = blockIdx.x;
  const int m = threadIdx.x;
  const int r = b * M_ + m;

  const bool mask = (run[r] != 0);

  // rank = clip(cumsum(mask) - 1, 0, M-1)
  int s1   = block_incl_scan(mask ? 1 : 0, m, sm);
  int rank = s1 - 1;
  rank     = rank < 0 ? 0 : rank;   // upper bound M-1 is automatic

  const float coeff = *coeff_p;
  const float pb    = *pb_p;
  float p = 0.f;
  if (mask) {
    const float x = dots[b * M_ + rank] + pb;
    p = coeff / (1.f + __expf(-x));              // sigmoid * coeff
  }
  const float ap  = acc_p[r];
  const bool  mc  = ((ap + p) < THRESH) && mask; // mask_continue
  const float upd = mc ? p : (mask ? (1.f - ap) : 0.f);

  rank_ws[r] = rank;
  upd_ws[r]  = upd;

  // left-pack destinations for still-running tokens
  int s2 = block_incl_scan(mc ? 1 : 0, m, sm);
  if (mc) srcmap[b * M_ + (s2 - 1)] = rank;      // packed source row in h
  if (m == M_ - 1) count_new[b] = s2;
}

// ---------------------------------------------------------------------------
// Kernel 3: fused elementwise emit (both outputs), float4 vectorized.
//   out[0 .. BMH)      = h_packed
//   out[BMH .. 2*BMH)  = weighted_h_new
// One 128-thread block per (b,m) row; each thread handles one float4 chunk.
// h re-reads hit the 192MB L2 (h is 64MB and was streamed by kernel 1).
// ---------------------------------------------------------------------------
__global__ __launch_bounds__(128) void act_emit(
    const float* __restrict__ h, const float* __restrict__ weighted_h,
    const float* __restrict__ pad_h, const int* __restrict__ rank_ws,
    const float* __restrict__ upd_ws, const int* __restrict__ srcmap,
    const int* __restrict__ count_new, float* __restrict__ out) {
  const int r = blockIdx.x;       // 0 .. B*M-1
  const int b = r >> 10;          // / M_
  const int m = r & (M_ - 1);     // % M_
  const int t = threadIdx.x;      // float4 chunk index, 0..127
  const size_t BMH = (size_t)BM_ * H_;

  // weighted_h_new = h_unp * update + weighted_h
  const float4 hv =
      ((const float4*)(h + (size_t)(b * M_ + rank_ws[r]) * H_))[t];
  const float4 wv = ((const float4*)(weighted_h + (size_t)r * H_))[t];
  const float  u  = upd_ws[r];
  float4 ov;
  ov.x = fmaf(hv.x, u, wv.x);
  ov.y = fmaf(hv.y, u, wv.y);
  ov.z = fmaf(hv.z, u, wv.z);
  ov.w = fmaf(hv.w, u, wv.w);
  ((float4*)(out + BMH + (size_t)r * H_))[t] = ov;

  // h_packed: gather still-running rows, pad tail with pad_h
  float4 pv;
  if (m < count_new[b]) {
    pv = ((const float4*)(h + (size_t)(b * M_ + srcmap[r]) * H_))[t];
  } else {
    pv = ((const float4*)pad_h)[t];
  }
  ((float4*)(out + (size_t)r * H_))[t] = pv;
}

// ---------------------------------------------------------------------------
extern "C" void kernel_launch(void* const* d_in, const int* in_sizes, int n_in,
                              void* d_out, int out_size, void* d_ws,
                              size_t ws_size, hipStream_t stream) {
  const float*   h          = (const float*)d_in[0];  // [B,M,H]
  const float*   coeff      = (const float*)d_in[1];  // scalar
  const float*   p_w        = (const float*)d_in[2];  // [1,H]
  const float*   p_b        = (const float*)d_in[3];  // [1]
  const float*   pad_h      = (const float*)d_in[4];  // [1,H]
  const float*   acc_p      = (const float*)d_in[5];  // [B,M,1]
  const float*   weighted_h = (const float*)d_in[6];  // [B,M,H]
  const uint8_t* run        = (const uint8_t*)d_in[7];// [B,M,1] bool (1B)

  float* out = (float*)d_out;

  // Workspace layout (all fully rewritten every call -> deterministic).
  char*  ws        = (char*)d_ws;
  float* dots      = (float*)(ws + 0);            // BM floats   (128 KB)
  int*   rank_ws   = (int*)(ws + 128 * 1024);     // BM ints     (128 KB)
  float* upd_ws    = (float*)(ws + 256 * 1024);   // BM floats   (128 KB)
  int*   srcmap    = (int*)(ws + 384 * 1024);     // BM ints     (128 KB)
  int*   count_new = (int*)(ws + 512 * 1024);     // B ints

  // K1: WMMA GEMV. 16 rows/wave, 8 waves/block -> 128 rows/block.
  act_gemv_wmma<<<BM_ / 128, 256, 0, stream>>>(h, p_w, dots);

  // K2: per-batch-row scans + halting update.
  act_scan<<<B_, M_, 0, stream>>>(run, acc_p, dots, coeff, p_b,
                                  rank_ws, upd_ws, srcmap, count_new);

  // K3: fused emit of both outputs.
  act_emit<<<BM_, 128, 0, stream>>>(h, weighted_h, pad_h, rank_ws, upd_ws,
                                    srcmap, count_new, out);
}